// Reg2HMLoss_75428215652867
// MI455X (gfx1250) — compile-verified
//
#include <hip/hip_runtime.h>
#include <hip/hip_bf16.h>

// Problem constants (match reference)
#define NB   512
#define KB   17
#define WW   48
#define HH   64
#define HW   (HH * WW)          // 3072
#define NK   (NB * KB)          // 8704
#define EPSF 1e-9f
#define INV_TOTAL (1.0f / 26738688.0f)   // 1 / (N*K*H*W)

typedef __attribute__((ext_vector_type(2))) float v2f;
typedef __attribute__((ext_vector_type(8))) float v8f;

// Exact fp32 sum of all 32 lane accumulators via V_WMMA_F32_16X16X4_F32.
// A: VGPR0 = acc, VGPR1 = 0  ->  A[m][0]=acc(lane m), A[m][2]=acc(lane m+16)
// B: all ones (layout-agnostic) ->  D[m][n] = acc(m) + acc(m+16)  (row sums, all cols)
// Lanes 0-15 hold rows 0-7 in d[0..7]; lanes 16-31 hold rows 8-15.
// EXEC must be all ones at the call site (uniform control flow, blockDim % 32 == 0).
__device__ __forceinline__ float wave_reduce_wmma(float acc) {
    v2f a; a.x = acc;  a.y = 0.0f;
    v2f b; b.x = 1.0f; b.y = 1.0f;
    v8f c = {};
    v8f d = __builtin_amdgcn_wmma_f32_16x16x4_f32(
        /*neg_a=*/false, a, /*neg_b=*/false, b,
        /*c_mod=*/(short)0, c, /*reuse_a=*/false, /*reuse_b=*/false);
    float s = ((d[0] + d[1]) + (d[2] + d[3])) + ((d[4] + d[5]) + (d[6] + d[7]));
    s += __shfl_xor(s, 16, 32);   // combine rows 0-7 half with rows 8-15 half
    return s;                     // full 32-lane sum in every lane
}

// One block per (n,k): build separable gaussian rows/cols in LDS (mask folded
// into gy), stream the 64x48 teacher tile as float4, accumulate squared diff,
// scale by tw^2 once, reduce through the matrix pipe.
__global__ __launch_bounds__(256)
void reg2hm_partials(const float* __restrict__ pred,
                     const float* __restrict__ sigma,
                     const float* __restrict__ teacher,
                     const float* __restrict__ twgt,
                     float* __restrict__ partial) {
    __shared__ float sgx[WW];
    __shared__ float sgy[HH];
    __shared__ float swave[8];

    const int b   = blockIdx.x;
    const int tid = threadIdx.x;

    // Per-(n,k) scalars (broadcast loads, L2-resident)
    const float mu_x = pred[2 * b + 0] * (float)WW;
    const float mu_y = pred[2 * b + 1] * (float)HH;
    const float sx   = sigma[2 * b + 0];
    const float sy   = sigma[2 * b + 1];
    const float tw0  = twgt[b];

    const bool vis = (mu_x - 3.0f * sx < (float)WW) &&
                     (mu_y - 3.0f * sy < (float)HH) &&
                     (mu_x + 3.0f * sx + 1.0f >= 0.0f) &&
                     (mu_y + 3.0f * sy + 1.0f >= 0.0f);
    const float mask = vis ? 1.0f : 0.0f;
    const float tw   = tw0 * mask;
    const float tw2  = tw * tw;          // hoisted: sum(tw*(gm-T))^2 = tw^2*sum(gm-T)^2

    const float inv_sx2 = 1.0f / (sx * sx + EPSF);
    const float inv_sy2 = 1.0f / (sy * sy + EPSF);

    // Separable gaussian: 112 exps per tile instead of 2 per element.
    // Fold the visibility mask into gy so the inner loop is pure FMA.
    if (tid < WW) {
        float dx = (float)tid - mu_x;
        sgx[tid] = __expf(-0.5f * dx * dx * inv_sx2);
    } else if (tid >= 64 && tid < 64 + HH) {
        int   h  = tid - 64;
        float dy = (float)h - mu_y;
        sgy[h]   = __expf(-0.5f * dy * dy * inv_sy2) * mask;   // target = g*mask
    }
    __syncthreads();

    // Stream the tile: 3072 floats = 768 float4, 256 threads -> 3 each.
    const float4* tp = (const float4*)(teacher + (size_t)b * HW);
    float acc = 0.0f;
#pragma unroll
    for (int j = 0; j < 3; ++j) {
        const int    q = tid + j * 256;   // float4 index, row-aligned (48 % 4 == 0)
        const float4 t = tp[q];
        const int    e = q * 4;
        const int    h = e / WW;
        const int    w = e - h * WW;
        const float gym = sgy[h];                      // mask pre-folded
        float d0 = fmaf(gym, sgx[w + 0], -t.x);
        float d1 = fmaf(gym, sgx[w + 1], -t.y);
        float d2 = fmaf(gym, sgx[w + 2], -t.z);
        float d3 = fmaf(gym, sgx[w + 3], -t.w);
        acc = fmaf(d0, d0, acc);
        acc = fmaf(d1, d1, acc);
        acc = fmaf(d2, d2, acc);
        acc = fmaf(d3, d3, acc);
    }
    acc *= tw2;   // apply weight once per thread

    // wave32 reduction through the matrix pipe, then 8 wave sums via LDS
    const float wsum = wave_reduce_wmma(acc);
    const int wave = tid >> 5;
    if ((tid & 31) == 0) swave[wave] = wsum;
    __syncthreads();
    if (tid == 0) {
        float s = 0.0f;
#pragma unroll
        for (int i = 0; i < 8; ++i) s += swave[i];
        partial[b] = s;
    }
}

// Single-block deterministic final reduction of the 8704 per-tile partials.
__global__ __launch_bounds__(256)
void reg2hm_finalize(const float* __restrict__ partial, float* __restrict__ out) {
    __shared__ float swave[8];
    const int tid = threadIdx.x;
    float acc = 0.0f;
    for (int i = tid; i < NK; i += 256) acc += partial[i];

    const float wsum = wave_reduce_wmma(acc);
    if ((tid & 31) == 0) swave[tid >> 5] = wsum;
    __syncthreads();
    if (tid == 0) {
        float s = 0.0f;
#pragma unroll
        for (int i = 0; i < 8; ++i) s += swave[i];
        out[0] = s * INV_TOTAL;   // mean * LOSS_WEIGHT(=1)
    }
}

extern "C" void kernel_launch(void* const* d_in, const int* in_sizes, int n_in,
                              void* d_out, int out_size, void* d_ws, size_t ws_size,
                              hipStream_t stream) {
    const float* pred    = (const float*)d_in[0];  // [N,K,2]
    const float* sigma   = (const float*)d_in[1];  // [N,K,2]
    const float* teacher = (const float*)d_in[2];  // [N,K,H,W]
    const float* twgt    = (const float*)d_in[3];  // [N,K,1]
    float* out = (float*)d_out;                    // scalar
    float* partial = (float*)d_ws;                 // NK floats of scratch

    reg2hm_partials<<<NK, 256, 0, stream>>>(pred, sigma, teacher, twgt, partial);
    reg2hm_finalize<<<1, 256, 0, stream>>>(partial, out);
}